// PyramidOccupancyNetworkIntermidiateFusion_29454885716405
// MI455X (gfx1250) — compile-verified
//
#include <hip/hip_runtime.h>

// Problem constants (from reference)
#define H_      200
#define W_      196
#define C_      256
#define NCAMS_  6
#define WH_     (W_ * H_)        // 39200 elements per camera-channel plane, p = w*H_ + h
#define N_      WH_              // number of BEV pixels, n = h*W_ + w
#define SS_     400
#define CELLS_  (SS_ * SS_)      // 160000 output cells
#define TILES_PER_CAM_ (WH_ / 16)          // 2450 (exact)
#define TOTAL_WAVES_   (NCAMS_ * TILES_PER_CAM_) // 14700

typedef float v2f __attribute__((ext_vector_type(2)));
typedef float v8f __attribute__((ext_vector_type(8)));

// ---------------------------------------------------------------------------
// Phase 0: zero the scatter workspace (must run every launch: harness does
// not re-poison/zero between graph replays).
// ---------------------------------------------------------------------------
__global__ void pon_zero_ws(float* __restrict__ ws, int n) {
    int i = blockIdx.x * blockDim.x + threadIdx.x;
    if (i < n) ws[i] = 0.0f;
}

// ---------------------------------------------------------------------------
// Phase 1: per-pixel channel reduction via V_WMMA_F32_16X16X4_F32, then
// scalar scatter-add into the 400x400 grid.
//   One wave32 handles one tile of 16 pixels (contiguous along H) for one
//   camera, accumulating all 256 channels in 64 K=4 WMMA steps.
// ---------------------------------------------------------------------------
__global__ __launch_bounds__(256) void pon_fuse_scatter(
    const float* __restrict__ td,     // [6][256][196][200] f32
    const float* __restrict__ mats,   // [6][4][4] f32
    const float* __restrict__ lc,     // [4][39200] f32
    const unsigned char* __restrict__ mask, // [200][196] bool, n = h*W + w
    const float* __restrict__ wcls,   // [256] f32
    float* __restrict__ ws_sum,       // [160000]
    float* __restrict__ ws_cnt)       // [160000]
{
    __shared__ float s_w[C_];
    const int tid = threadIdx.x;
    s_w[tid] = wcls[tid];             // blockDim.x == 256
    __syncthreads();

    const int lane  = tid & 31;
    const int gwave = blockIdx.x * (blockDim.x >> 5) + (tid >> 5);
    if (gwave >= TOTAL_WAVES_) return;

    const int cam  = gwave / TILES_PER_CAM_;
    const int tile = gwave - cam * TILES_PER_CAM_;

    // A-matrix (16x4 f32) lane layout: row M = lane%16 (pixel),
    // VGPR0/1 hold K = koff, koff+1 with koff = 2*(lane/16).
    const int pix  = tile * 16 + (lane & 15);   // plane-linear index p = w*H + h
    const int koff = (lane >> 4) << 1;          // 0 or 2

    const float* __restrict__ Ap = td + (size_t)cam * C_ * WH_ + pix;

    v8f acc0 = {}; v8f acc1 = {}; v8f acc2 = {}; v8f acc3 = {};
    for (int kc = 0; kc < C_; kc += 16) {
        {
            const int k = kc + koff;
            v2f a = { Ap[(size_t)k * WH_], Ap[(size_t)(k + 1) * WH_] };
            v2f b = { s_w[k], s_w[k + 1] };   // broadcast over all 16 B columns
            acc0 = __builtin_amdgcn_wmma_f32_16x16x4_f32(false, a, false, b,
                                                         (short)0, acc0, false, false);
        }
        {
            const int k = kc + 4 + koff;
            v2f a = { Ap[(size_t)k * WH_], Ap[(size_t)(k + 1) * WH_] };
            v2f b = { s_w[k], s_w[k + 1] };
            acc1 = __builtin_amdgcn_wmma_f32_16x16x4_f32(false, a, false, b,
                                                         (short)0, acc1, false, false);
        }
        {
            const int k = kc + 8 + koff;
            v2f a = { Ap[(size_t)k * WH_], Ap[(size_t)(k + 1) * WH_] };
            v2f b = { s_w[k], s_w[k + 1] };
            acc2 = __builtin_amdgcn_wmma_f32_16x16x4_f32(false, a, false, b,
                                                         (short)0, acc2, false, false);
        }
        {
            const int k = kc + 12 + koff;
            v2f a = { Ap[(size_t)k * WH_], Ap[(size_t)(k + 1) * WH_] };
            v2f b = { s_w[k], s_w[k + 1] };
            acc3 = __builtin_amdgcn_wmma_f32_16x16x4_f32(false, a, false, b,
                                                         (short)0, acc3, false, false);
        }
    }
    v8f acc = (acc0 + acc1) + (acc2 + acc3);

    // D layout (16x16 f32, columns identical due to broadcast-B):
    //   lane L, component v  ->  pixel  v + 8*(L/16)
    // Extract one pixel per scattering lane with a 7-select tree.
    const int k3 = lane & 7;
    float s01 = (k3 & 1) ? acc[1] : acc[0];
    float s23 = (k3 & 1) ? acc[3] : acc[2];
    float s45 = (k3 & 1) ? acc[5] : acc[4];
    float s67 = (k3 & 1) ? acc[7] : acc[6];
    float sA  = (k3 & 2) ? s23 : s01;
    float sB  = (k3 & 2) ? s67 : s45;
    float sv  = (k3 & 4) ? sB  : sA;

    // Lanes 0-7 scatter pixels 0-7; lanes 16-23 scatter pixels 8-15.
    if ((lane & 15) < 8) {
        const int pin = k3 + ((lane >> 4) << 3);  // 0..15 within tile
        const int p   = tile * 16 + pin;          // plane-linear (w*H + h)
        const int h   = p % H_;
        const int w   = p / H_;
        const int n   = h * W_ + w;               // reference pixel ordering
        if (mask[n]) {
            const float gx = lc[n];
            const float gy = lc[N_ + n];
            const float* __restrict__ M = mats + cam * 16;
            const float x = M[0] * gx + M[1] * gy + M[3];
            const float y = M[4] * gx + M[5] * gy + M[7];
            const int i0 = (int)__builtin_rintf((x + 100.0f) * 2.0f);
            const int i1 = (int)__builtin_rintf((y + 100.0f) * 2.0f);
            if (i0 >= 0 && i0 < SS_ && i1 >= 0 && i1 < SS_) {
                const int cell = i0 * SS_ + i1;
                atomicAdd(ws_sum + cell, sv);
                atomicAdd(ws_cnt + cell, 1.0f);
            }
        }
    }
}

// ---------------------------------------------------------------------------
// Phase 2: finalize — divide by count (cells with cnt==0 have sum==0) + bias.
// ---------------------------------------------------------------------------
__global__ void pon_finalize(const float* __restrict__ ws_sum,
                             const float* __restrict__ ws_cnt,
                             const float* __restrict__ b_cls,
                             float* __restrict__ out)
{
    int i = blockIdx.x * blockDim.x + threadIdx.x;
    if (i < CELLS_) {
        float c = ws_cnt[i];
        float s = ws_sum[i];
        out[i] = ((c >= 1.0f) ? (s / c) : s) + b_cls[0];
    }
}

extern "C" void kernel_launch(void* const* d_in, const int* in_sizes, int n_in,
                              void* d_out, int out_size, void* d_ws, size_t ws_size,
                              hipStream_t stream) {
    const float*         td   = (const float*)d_in[0];          // td_feats
    const float*         mats = (const float*)d_in[1];          // cam_to_ego
    const float*         lc   = (const float*)d_in[2];          // logits_coordinates
    const unsigned char* mask = (const unsigned char*)d_in[3];  // single_mask (bool)
    const float*         wcls = (const float*)d_in[4];          // w_cls
    const float*         bcls = (const float*)d_in[5];          // b_cls (scalar)
    float* out = (float*)d_out;

    float* ws_sum = (float*)d_ws;
    float* ws_cnt = ws_sum + CELLS_;

    pon_zero_ws<<<(2 * CELLS_ + 255) / 256, 256, 0, stream>>>(ws_sum, 2 * CELLS_);

    const int blocks = (TOTAL_WAVES_ + 7) / 8;   // 8 waves (256 threads) per block
    pon_fuse_scatter<<<blocks, 256, 0, stream>>>(td, mats, lc, mask, wcls,
                                                 ws_sum, ws_cnt);

    pon_finalize<<<(CELLS_ + 255) / 256, 256, 0, stream>>>(ws_sum, ws_cnt, bcls, out);
}